// FixedPointBackprop_36395552866555
// MI455X (gfx1250) — compile-verified
//
#include <hip/hip_runtime.h>

// FixedPointBackprop dLdx (solve mode, symmetric), b=512, m=64, n=1.
// Per batch: g = normalize(x@y); H = I - (x - g (g^T x))/||xy||;
//            z = (H H^T)^{-1} (H dLdy); w = (z - g (g.z))/||xy||;
//            dLdx = 0.5 (w y^T + y w^T).
// H H^T (64x64x64 f32 GEMM) via V_WMMA_F32_16X16X4_F32 tiles (CDNA5 fp32 WMMA).

#define MDIM 64
#define STR  68          // LDS row stride (pad vs 64-bank conflicts)
#define EPSN 1e-12f
#define NTHR 128

typedef __attribute__((ext_vector_type(2))) float v2f;
typedef __attribute__((ext_vector_type(8))) float v8f;

__launch_bounds__(NTHR)
__global__ void fixedpoint_bwd_kernel(const float* __restrict__ x,
                                      const float* __restrict__ y,
                                      const float* __restrict__ dLdy,
                                      float* __restrict__ out)
{
    __shared__ float Xs[MDIM * STR];   // x, 64x64
    __shared__ float Hs[MDIM * STR];   // H, 64x64
    __shared__ float As[MDIM * STR];   // HHT -> elimination workspace
    __shared__ float yv[MDIM], qv[MDIM], xyv[MDIM], gv[MDIM], uv[MDIM];
    __shared__ float rhs[MDIM], zs[MDIM], wvec[MDIM], red[MDIM];
    __shared__ float s_nrm, s_gz;

    const int b    = blockIdx.x;
    const int tid  = threadIdx.x;
    const int lane = tid & 31;
    const int wid  = tid >> 5;        // wave 0..3
    const int half = lane >> 4;       // 0/1 (K split per ISA A/B layout)
    const int l16  = lane & 15;

    const float* xb = x    + (size_t)b * MDIM * MDIM;
    const float* yb = y    + (size_t)b * MDIM;
    const float* qb = dLdy + (size_t)b * MDIM;
    float*       ob = out  + (size_t)b * MDIM * MDIM;

    // ---- stage inputs in LDS (coalesced) ----
    for (int e = tid; e < MDIM * MDIM; e += NTHR)
        Xs[(e >> 6) * STR + (e & 63)] = xb[e];
    if (tid < MDIM) { yv[tid] = yb[tid]; qv[tid] = qb[tid]; }
    __syncthreads();

    // xy = x @ y (one row-dot per thread)
    if (tid < MDIM) {
        float acc = 0.0f;
        for (int j = 0; j < MDIM; ++j) acc += Xs[tid * STR + j] * yv[j];
        xyv[tid] = acc;
        red[tid] = acc * acc;
    }
    __syncthreads();
    // tree-reduce ||xy||^2 over 64 entries
    #pragma unroll
    for (int s = 32; s > 0; s >>= 1) {
        if (tid < s) red[tid] += red[tid + s];
        __syncthreads();
    }
    if (tid == 0) s_nrm = fmaxf(sqrtf(red[0]), EPSN);
    __syncthreads();
    const float inv_nrm = 1.0f / s_nrm;
    if (tid < MDIM) gv[tid] = xyv[tid] * inv_nrm;
    __syncthreads();

    // u = x^T g (one column-dot per thread)
    if (tid < MDIM) {
        float acc = 0.0f;
        for (int i = 0; i < MDIM; ++i) acc += gv[i] * Xs[i * STR + tid];
        uv[tid] = acc;
    }
    __syncthreads();

    // H = I - (x - g u^T)/nrm
    for (int e = tid; e < MDIM * MDIM; e += NTHR) {
        const int i = e >> 6, j = e & 63;
        Hs[i * STR + j] = ((i == j) ? 1.0f : 0.0f)
                        - (Xs[i * STR + j] - gv[i] * uv[j]) * inv_nrm;
    }
    __syncthreads();

    // ---- HHT = H H^T via V_WMMA_F32_16X16X4_F32 ----
    // Wave `wid` owns row-band [16*wid,16*wid+16). A: 16x4 of H; B: 4x16 of
    // H^T (rows of H for the N-tile). ISA layout: lanes 0-15 K={k0,k0+1},
    // lanes 16-31 K={k0+2,k0+3}. All 128 threads active (EXEC all-ones).
    const int arow = 16 * wid + l16;
    for (int tn = 0; tn < 4; ++tn) {
        const int brow = 16 * tn + l16;
        v8f acc = {};
        #pragma unroll
        for (int k0 = 0; k0 < MDIM; k0 += 4) {
            const int kk = k0 + 2 * half;
            v2f a, bf;
            a.x  = Hs[arow * STR + kk];
            a.y  = Hs[arow * STR + kk + 1];
            bf.x = Hs[brow * STR + kk];
            bf.y = Hs[brow * STR + kk + 1];
            acc = __builtin_amdgcn_wmma_f32_16x16x4_f32(
                false, a, false, bf, (short)0, acc, false, false);
        }
        // D layout: vgpr r, lanes 0-15 -> M=r, N=lane; lanes 16-31 -> M=r+8.
        #pragma unroll
        for (int r = 0; r < 8; ++r)
            As[(16 * wid + r + 8 * half) * STR + 16 * tn + l16] = acc[r];
    }
    __syncthreads();

    // rhs = H @ dLdy
    if (tid < MDIM) {
        float acc = 0.0f;
        for (int k = 0; k < MDIM; ++k) acc += Hs[tid * STR + k] * qv[k];
        rhs[tid] = acc;
    }
    __syncthreads();

    // ---- solve HHT z = rhs: no-pivot Gaussian elimination (SPD) ----
    // One barrier per round. Column k and row k are read-only within round k
    // (we never write the eliminated column; back-substitution only reads the
    // upper triangle + diagonal), so each thread recomputes pivinv and its
    // factor locally — no broadcast phase. Trailing update is element-parallel.
    for (int k = 0; k < MDIM; ++k) {
        const float pivinv = 1.0f / As[k * STR + k];
        if (tid > k && tid < MDIM)
            rhs[tid] -= As[tid * STR + k] * pivinv * rhs[k];
        const int span = MDIM - 1 - k;               // rows/cols k+1..63
        for (int e = tid; e < span * span; e += NTHR) {
            const int i = k + 1 + e / span;
            const int j = k + 1 + e % span;
            As[i * STR + j] -= As[i * STR + k] * pivinv * As[k * STR + j];
        }
        __syncthreads();
    }
    // Back-substitution, one barrier per round: rhs[i] is only ever written by
    // thread i itself, so z[i] = rhs[i]/A[i][i] needs no barrier before it;
    // only the z[i] broadcast to the column-sweep does.
    for (int i = MDIM - 1; i >= 0; --i) {
        if (tid == i) zs[i] = rhs[i] / As[i * STR + i];
        __syncthreads();
        if (tid < i) rhs[tid] -= As[tid * STR + i] * zs[i];
    }
    __syncthreads();

    // gz = g . z (tree reduction), then w = (z - g*gz)/nrm
    if (tid < MDIM) red[tid] = gv[tid] * zs[tid];
    __syncthreads();
    #pragma unroll
    for (int s = 32; s > 0; s >>= 1) {
        if (tid < s) red[tid] += red[tid + s];
        __syncthreads();
    }
    if (tid == 0) s_gz = red[0];
    __syncthreads();
    if (tid < MDIM) wvec[tid] = (zs[tid] - gv[tid] * s_gz) * inv_nrm;
    __syncthreads();

    // dLdx = 0.5 (w y^T + y w^T), coalesced store
    for (int e = tid; e < MDIM * MDIM; e += NTHR) {
        const int i = e >> 6, j = e & 63;
        ob[e] = 0.5f * (wvec[i] * yv[j] + yv[i] * wvec[j]);
    }
}

extern "C" void kernel_launch(void* const* d_in, const int* in_sizes, int n_in,
                              void* d_out, int out_size, void* d_ws, size_t ws_size,
                              hipStream_t stream) {
    const float* x    = (const float*)d_in[0];
    const float* y    = (const float*)d_in[1];
    // d_in[2] = lambds, unused by the reference's dLdx_fnc.
    const float* dLdy = (const float*)d_in[3];
    float* out = (float*)d_out;

    const int b = in_sizes[0] / (MDIM * MDIM);   // 512
    fixedpoint_bwd_kernel<<<dim3(b), dim3(NTHR), 0, stream>>>(x, y, dLdy, out);
}